// ModelNew_1580547969507
// MI455X (gfx1250) — compile-verified
//
#include <hip/hip_runtime.h>
#include <stdint.h>

typedef __attribute__((ext_vector_type(2))) float v2f;
typedef __attribute__((ext_vector_type(8))) float v8f;
typedef __attribute__((ext_vector_type(4))) unsigned int v4u;
typedef __attribute__((ext_vector_type(4))) int v4i;
typedef __attribute__((ext_vector_type(8))) int v8i;

#define Bc 32
#define Lc 8192
#define Fc 64
#define Hc 64
#define CHUNKS 8
#define ROWS_PER_WG (Lc / CHUNKS)            // 1024
#define WAVES 8
#define THREADS (WAVES * 32)                 // 256
#define TILE_ROWS (WAVES * 16)               // 128 rows per TDM tile
#define PASSES (ROWS_PER_WG / TILE_ROWS)     // 8
#define PITCH 66                             // 64 data dwords + 2 pad dwords/row
#define YT_FLOATS (Hc * PITCH)               // 4224
#define XT_FLOATS (TILE_ROWS * PITCH)        // 8448

#if __has_builtin(__builtin_amdgcn_tensor_load_to_lds)
#define HAVE_TDM 1
#else
#define HAVE_TDM 0
#endif

#if HAVE_TDM
// Build a Tensor DMA Descriptor (D#, ISA ch.8) for one 128x64 f32 tile:
// 2D tile, data_size=4B, LDS padding of 2 dwords after every 64 dwords
// (row pitch 66 -> bank-conflict-free, 8B-aligned fragment reads).
__device__ __forceinline__ void tdm_load_x_tile(const float* gsrc, unsigned lds_byte_off)
{
    const unsigned long long ga = (unsigned long long)(uintptr_t)gsrc;
    v4u g0;
    g0[0] = 1u;                                    // count=1 (valid user D#)
    g0[1] = lds_byte_off;                          // lds_addr  (bits 63:32)
    g0[2] = (unsigned)ga;                          // global_addr[31:0]
    g0[3] = (unsigned)((ga >> 32) & 0x01FFFFFFu)   // global_addr[56:32]
          | (2u << 30);                            // type=2 ("image")
    v8i g1;
    g1[0] = (int)((2u << 16)                       // data_size: 4 bytes
                | (1u << 20)                       // pad_enable
                | (5u << 22)                       // pad_interval: 64 dwords
                | (1u << 25));                     // pad_amount: 2 dwords
    g1[1] = (int)(((unsigned)Fc & 0xFFFFu) << 16); // tensor_dim0 lo16 = 64
    g1[2] = (int)(((unsigned)Lc & 0xFFFFu) << 16); // dim0 hi16=0 | tensor_dim1 lo16 = 8192
    g1[3] = (int)(((unsigned)Fc) << 16);           // dim1 hi16=0 | tile_dim0 = 64
    g1[4] = (int)TILE_ROWS;                        // tile_dim1 = 128 | tile_dim2 = 0
    g1[5] = (int)Fc;                               // tensor_dim0_stride lo32 = 64
    g1[6] = 0;                                     // stride0 hi16 | stride1 lo16
    g1[7] = 0;                                     // stride1 hi32
    v4i gz = {0, 0, 0, 0};                         // groups 2/3 unused (2D tile)
#if defined(__clang_major__) && (__clang_major__ >= 23)
    v8i gz8 = {0, 0, 0, 0, 0, 0, 0, 0};
    __builtin_amdgcn_tensor_load_to_lds(g0, g1, gz, gz, gz8, 0);
#else
    __builtin_amdgcn_tensor_load_to_lds(g0, g1, gz, gz, 0);
#endif
}
#endif

__global__ __launch_bounds__(THREADS)
void gemm_relu_sum_kernel(const float* __restrict__ x,
                          const float* __restrict__ y,
                          float* __restrict__ partial)
{
    __shared__ __align__(16) float yt[YT_FLOATS];     // Y[b]^T, padded pitch 66
    __shared__ __align__(16) float xt[2][XT_FLOATS];  // double-buffered X tiles
    __shared__ float red[THREADS];

    const int tid   = threadIdx.x;
    const int wave  = tid >> 5;
    const int lane  = tid & 31;
    const int b     = blockIdx.x / CHUNKS;
    const int chunk = blockIdx.x % CHUNKS;

    // ---- stage Y[b] (64x64 f32) transposed into LDS, coalesced ----
    const float* yb = y + (size_t)b * Fc * Hc;
    #pragma unroll
    for (int idx = tid; idx < Fc * Hc; idx += THREADS) {
        const int f = idx >> 6;
        const int h = idx & 63;
        yt[h * PITCH + f] = yb[idx];
    }

    const float* xchunk = x + ((size_t)b * Lc + (size_t)chunk * ROWS_PER_WG) * Fc;

    // WMMA f32 16x16x4 striping: lanes 0-15 -> {K,K+1}, lanes 16-31 -> {K+2,K+3}
    const int lhalf = lane & 15;
    const int khalf = (lane >> 4) * 2;
    float ssum = 0.f;

#if HAVE_TDM
    // ---- prologue: TDM streams tile 0 into xt[0] ----
    if (wave == 0) {
        tdm_load_x_tile(xchunk, (unsigned)(uintptr_t)&xt[0][0]);
        __builtin_amdgcn_s_wait_tensorcnt(0);
    }
    __syncthreads();                       // yt + xt[0] visible to all waves

    for (int p = 0; p < PASSES; ++p) {
        const int cur = p & 1;
        // Kick DMA for next tile while we compute this one (TENSORcnt overlap).
        if (wave == 0 && (p + 1) < PASSES)
            tdm_load_x_tile(xchunk + (size_t)(p + 1) * TILE_ROWS * Fc,
                            (unsigned)(uintptr_t)&xt[cur ^ 1][0]);

        // A fragments for this wave's 16x64 tile, from padded LDS (ds_load_b64).
        const float* ap = &xt[cur][(wave * 16 + lhalf) * PITCH + khalf];
        v2f a[16];
        #pragma unroll
        for (int i = 0; i < 16; ++i) a[i] = *(const v2f*)(ap + i * 4);

        #pragma unroll
        for (int nt = 0; nt < 4; ++nt) {
            const float* bp = &yt[(nt * 16 + lhalf) * PITCH + khalf];
            v8f acc = {0.f, 0.f, 0.f, 0.f, 0.f, 0.f, 0.f, 0.f};
            #pragma unroll
            for (int i = 0; i < 16; ++i) {
                v2f bfrag = *(const v2f*)(bp + i * 4);
                acc = __builtin_amdgcn_wmma_f32_16x16x4_f32(
                    false, a[i], false, bfrag, (short)0, acc, false, false);
            }
            #pragma unroll
            for (int j = 0; j < 8; ++j)
                ssum += fmaxf(acc[j], 0.f);
        }

        __syncthreads();                   // all waves done reading xt[cur]
        if ((p + 1) < PASSES) {
            if (wave == 0) __builtin_amdgcn_s_wait_tensorcnt(0);
            __syncthreads();               // xt[cur^1] ready for everyone
        }
    }
#else
    // ---- fallback: direct global A-fragment loads (round-1 path) ----
    __syncthreads();
    for (int p = 0; p < PASSES; ++p) {
        const int m0 = p * TILE_ROWS + wave * 16;
        const float* rowp = xchunk + (size_t)(m0 + lhalf) * Fc + khalf;
        if (p + 1 < PASSES)
            __builtin_prefetch(rowp + (size_t)TILE_ROWS * Fc, 0, 3);
        v2f a[16];
        #pragma unroll
        for (int i = 0; i < 16; ++i) a[i] = *(const v2f*)(rowp + i * 4);
        #pragma unroll
        for (int nt = 0; nt < 4; ++nt) {
            const float* bp = &yt[(nt * 16 + lhalf) * PITCH + khalf];
            v8f acc = {0.f, 0.f, 0.f, 0.f, 0.f, 0.f, 0.f, 0.f};
            #pragma unroll
            for (int i = 0; i < 16; ++i) {
                v2f bfrag = *(const v2f*)(bp + i * 4);
                acc = __builtin_amdgcn_wmma_f32_16x16x4_f32(
                    false, a[i], false, bfrag, (short)0, acc, false, false);
            }
            #pragma unroll
            for (int j = 0; j < 8; ++j)
                ssum += fmaxf(acc[j], 0.f);
        }
    }
#endif

    // ---- workgroup reduction ----
    red[tid] = ssum;
    __syncthreads();
    #pragma unroll
    for (int s = THREADS / 2; s > 0; s >>= 1) {
        if (tid < s) red[tid] += red[tid + s];
        __syncthreads();
    }
    if (tid == 0) partial[blockIdx.x] = red[0];
}

__global__ __launch_bounds__(64)
void finalize_kernel(const float* __restrict__ partial, float* __restrict__ out)
{
    __shared__ float sm[Bc];
    const int t = threadIdx.x;
    if (t < Bc) {
        float s = 0.f;
        #pragma unroll
        for (int c = 0; c < CHUNKS; ++c) s += partial[t * CHUNKS + c];
        sm[t] = s / (float)Lc;
    }
    __syncthreads();
    if (t == 0) {
        float m = sm[0];
        #pragma unroll
        for (int i = 1; i < Bc; ++i) m = fmaxf(m, sm[i]);
        out[0] = m;
    }
}

extern "C" void kernel_launch(void* const* d_in, const int* in_sizes, int n_in,
                              void* d_out, int out_size, void* d_ws, size_t ws_size,
                              hipStream_t stream)
{
    const float* x = (const float*)d_in[0];   // (B, L, F) f32
    const float* y = (const float*)d_in[1];   // (B, F, H) f32
    float* out     = (float*)d_out;           // scalar f32
    float* partial = (float*)d_ws;            // Bc*CHUNKS = 256 floats

    gemm_relu_sum_kernel<<<Bc * CHUNKS, THREADS, 0, stream>>>(x, y, partial);
    finalize_kernel<<<1, 64, 0, stream>>>(partial, out);
}